// TCTripletLoss_4896262717964
// MI455X (gfx1250) — compile-verified
//
#include <hip/hip_runtime.h>
#include <hip/hip_bf16.h>

typedef __attribute__((ext_vector_type(2))) float v2f;
typedef __attribute__((ext_vector_type(8))) float v8f;

#define NPATCH 128
#define DIM    512
#define BSZ    32
#define ALPHAF 1.0f
#define EPSF   1e-9f

// ---------------------------------------------------------------------------
// Stage 1: row-wise L2 normalization of latent (8192 x 512 fp32).
// ---------------------------------------------------------------------------
__global__ void tl_normalize(const float* __restrict__ in, float* __restrict__ out) {
    const int row = blockIdx.x;          // 0..8191
    const int tid = threadIdx.x;         // 0..127
    const float4* src = (const float4*)(in + (size_t)row * DIM);
    float4 v = src[tid];
    float ss = v.x * v.x + v.y * v.y + v.z * v.z + v.w * v.w;
    #pragma unroll
    for (int off = 16; off > 0; off >>= 1) ss += __shfl_xor(ss, off, 32);
    __shared__ float wsum[4];
    const int wave = tid >> 5;
    if ((tid & 31) == 0) wsum[wave] = ss;
    __syncthreads();
    const float tot = wsum[0] + wsum[1] + wsum[2] + wsum[3];
    const float inv = rsqrtf(tot);
    float4 o;
    o.x = v.x * inv; o.y = v.y * inv; o.z = v.z * inv; o.w = v.w * inv;
    ((float4*)(out + (size_t)row * DIM))[tid] = o;
}

// ---------------------------------------------------------------------------
// Stage 2: Gram matrices with V_WMMA_F32_16X16X4_F32, software-pipelined.
// Each wave: one 16x128 strip (8 accumulators = 64 VGPRs), K=512 step 4.
// Ping-pong fragment sets overlap the 9-load clause with the 8 WMMAs.
// __launch_bounds__(128, 1) lifts the 128-VGPR occupancy cap that caused
// scratch spilling of accumulators in the previous build (~110 live VGPRs
// needed; allocator may now use up to 256).
// ---------------------------------------------------------------------------
__global__ void __launch_bounds__(128, 1)
tl_gram(const float* __restrict__ nrm, float* __restrict__ G) {
    const int lane  = threadIdx.x & 31;
    const int wave  = threadIdx.x >> 5;
    const int ntile = blockIdx.x * 4 + wave;   // 0..7
    const int gram  = blockIdx.y;              // 0:tt 1:cc 2:tc
    const int b     = blockIdx.z;              // 0..31

    const float* X = nrm + ((size_t)b * 2 * NPATCH + ((gram == 1) ? NPATCH : 0)) * DIM;
    const float* Y = nrm + ((size_t)b * 2 * NPATCH + ((gram == 0) ? 0 : NPATCH)) * DIM;
    float* Gout = G + ((size_t)gram * BSZ + b) * (NPATCH * NPATCH);

    const int r     = lane & 15;               // row within tile
    const int khalf = (lane >> 4) << 1;        // 0 or 2
    const float* Abase = X + (size_t)(ntile * 16 + r) * DIM + khalf;
    const float* Bbase = Y + (size_t)r * DIM + khalf;   // + mt*16*DIM as imm offset

    v8f acc[8];
    #pragma unroll
    for (int mt = 0; mt < 8; ++mt) {
        v8f z = {0.f, 0.f, 0.f, 0.f, 0.f, 0.f, 0.f, 0.f};
        acc[mt] = z;
    }

    // Prologue: fragment set A holds k-step 0.
    v2f aA = *(const v2f*)(Abase);
    v2f bA[8];
    #pragma unroll
    for (int mt = 0; mt < 8; ++mt)
        bA[mt] = *(const v2f*)(Bbase + (size_t)mt * (16 * DIM));

    // Steady state: steps 0..125 (63 iterations, 2 steps each).
    for (int kk = 0; kk < DIM / 4 - 2; kk += 2) {
        v2f aB = *(const v2f*)(Abase + (kk + 1) * 4);
        v2f bB[8];
        #pragma unroll
        for (int mt = 0; mt < 8; ++mt)
            bB[mt] = *(const v2f*)(Bbase + (size_t)mt * (16 * DIM) + (kk + 1) * 4);
        #pragma unroll
        for (int mt = 0; mt < 8; ++mt)
            acc[mt] = __builtin_amdgcn_wmma_f32_16x16x4_f32(
                false, aA, false, bA[mt], (short)0, acc[mt], false, false);

        aA = *(const v2f*)(Abase + (kk + 2) * 4);
        #pragma unroll
        for (int mt = 0; mt < 8; ++mt)
            bA[mt] = *(const v2f*)(Bbase + (size_t)mt * (16 * DIM) + (kk + 2) * 4);
        #pragma unroll
        for (int mt = 0; mt < 8; ++mt)
            acc[mt] = __builtin_amdgcn_wmma_f32_16x16x4_f32(
                false, aB, false, bB[mt], (short)0, acc[mt], false, false);
    }

    // Epilogue: set A holds step 126; load and run step 127.
    {
        v2f aB = *(const v2f*)(Abase + (DIM / 4 - 1) * 4);
        v2f bB[8];
        #pragma unroll
        for (int mt = 0; mt < 8; ++mt)
            bB[mt] = *(const v2f*)(Bbase + (size_t)mt * (16 * DIM) + (DIM / 4 - 1) * 4);
        #pragma unroll
        for (int mt = 0; mt < 8; ++mt)
            acc[mt] = __builtin_amdgcn_wmma_f32_16x16x4_f32(
                false, aA, false, bA[mt], (short)0, acc[mt], false, false);
        #pragma unroll
        for (int mt = 0; mt < 8; ++mt)
            acc[mt] = __builtin_amdgcn_wmma_f32_16x16x4_f32(
                false, aB, false, bB[mt], (short)0, acc[mt], false, false);
    }

    // D layout: lane L, VGPR v -> row (v + (L>=16?8:0)), col (L%16).
    const int rowbase = ntile * 16 + ((lane >= 16) ? 8 : 0);
    const int col     = lane & 15;
    #pragma unroll
    for (int mt = 0; mt < 8; ++mt) {
        #pragma unroll
        for (int v = 0; v < 8; ++v) {
            Gout[(size_t)(rowbase + v) * NPATCH + mt * 16 + col] = acc[mt][v];
        }
    }
}

// ---------------------------------------------------------------------------
// Stage 3: triplet reduction.
// ppn = alpha - 2*Gpp[n,m] + 2*H[n,k];  num/den = sum_{v>0} v / (cnt(v>0)+eps)
// One block per (call, b, n): 128 threads (one per m), H row cached in LDS.
// ---------------------------------------------------------------------------
__global__ void tl_triplet(const float* __restrict__ G, float* __restrict__ partial) {
    const int x    = blockIdx.x;          // 0..8191
    const int call = x >> 12;             // 0,1
    const int b    = (x >> 7) & 31;
    const int n    = x & 127;
    const int m    = threadIdx.x;         // 0..127

    const float* Gpp = G + ((size_t)call * BSZ + b) * (NPATCH * NPATCH);
    const float* Gtc = G + ((size_t)2 * BSZ + b) * (NPATCH * NPATCH);

    __shared__ float Hrow[NPATCH];
    Hrow[m] = (call == 0) ? Gtc[(size_t)n * NPATCH + m]
                          : Gtc[(size_t)m * NPATCH + n];
    __syncthreads();

    const float s = ALPHAF - 2.0f * Gpp[(size_t)n * NPATCH + m];
    float sum = 0.0f, cnt = 0.0f;
    #pragma unroll 8
    for (int k = 0; k < NPATCH; ++k) {
        const float v = fmaf(2.0f, Hrow[k], s);
        if (v > 0.0f) { sum += v; cnt += 1.0f; }
    }
    float r = sum / (cnt + EPSF);

    #pragma unroll
    for (int off = 16; off > 0; off >>= 1) r += __shfl_xor(r, off, 32);
    __shared__ float wsum[4];
    if ((m & 31) == 0) wsum[m >> 5] = r;
    __syncthreads();
    if (m == 0) partial[x] = wsum[0] + wsum[1] + wsum[2] + wsum[3];
}

// ---------------------------------------------------------------------------
// Stage 4: deterministic final reduction; loss = (S0 + S1) / (32*128*128).
// ---------------------------------------------------------------------------
__global__ void tl_finalize(const float* __restrict__ partial, float* __restrict__ out) {
    const int tid = threadIdx.x;          // 256 threads
    float s = 0.0f;
    for (int i = tid; i < 8192; i += 256) s += partial[i];
    #pragma unroll
    for (int off = 16; off > 0; off >>= 1) s += __shfl_xor(s, off, 32);
    __shared__ float wsum[8];
    if ((tid & 31) == 0) wsum[tid >> 5] = s;
    __syncthreads();
    if (tid == 0) {
        float t = 0.0f;
        #pragma unroll
        for (int i = 0; i < 8; ++i) t += wsum[i];
        out[0] = t * (1.0f / (float)(BSZ * NPATCH * NPATCH));
    }
}

// ---------------------------------------------------------------------------
extern "C" void kernel_launch(void* const* d_in, const int* in_sizes, int n_in,
                              void* d_out, int out_size, void* d_ws, size_t ws_size,
                              hipStream_t stream) {
    const float* latent = (const float*)d_in[0];     // 8192*512 fp32
    float* ws      = (float*)d_ws;
    float* nrm     = ws;                                    // 8192*512      (16 MB)
    float* G       = nrm + (size_t)8192 * 512;              // 3*32*128*128  ( 6 MB)
    float* partial = G + (size_t)3 * BSZ * NPATCH * NPATCH; // 8192 floats
    float* out     = (float*)d_out;

    tl_normalize<<<8192, 128, 0, stream>>>(latent, nrm);

    dim3 g2(2, 3, BSZ);                 // 2 blocks x 4 waves -> 8 n-tiles; 3 grams; 32 batches
    tl_gram<<<g2, 128, 0, stream>>>(nrm, G);

    tl_triplet<<<8192, 128, 0, stream>>>(G, partial);   // (2 calls x 32 b x 128 n)

    tl_finalize<<<1, 256, 0, stream>>>(partial, out);
}